// Tri_Att_GCL_32933809225923
// MI455X (gfx1250) — compile-verified
//
#include <hip/hip_runtime.h>
#include <hip/hip_bf16.h>

// ---------------------------------------------------------------------------
// Tri-attention GCL for MI455X (gfx1250, wave32).
// P layout per edge (NCH=528 f32): [q 4x32][k 4x32][v 4x32][g 4x32][b 4][pad 12]
// so neighbor gathers of k and v are each one contiguous 512B region.
// All GEMMs use v_wmma_f32_16x16x4_f32 (f32 in/out: compute is free, we are
// gather-bandwidth-bound; ~3.2GB gather / 23.3TB/s dominates).
// ---------------------------------------------------------------------------

#define E_TOTAL  200000
#define IN_DIM   128
#define HIDC     32
#define NHEAD    4
#define OUT_DIM  128
#define MAXK     16
#define NCH      528            // 33 tiles of 16
#define NTILE_P  33
#define TILE_M   16
#define NBLK     (E_TOTAL / TILE_M)   // 12500 (E divisible by 16)

typedef __attribute__((ext_vector_type(2))) float v2f;
typedef __attribute__((ext_vector_type(8))) float v8f;

// 1-ulp hardware reciprocal (v_rcp_f32) — plenty for sigmoid/softmax here.
__device__ __forceinline__ float fast_rcp(float x) {
  return __builtin_amdgcn_rcpf(x);
}

// ---------------------------------------------------------------------------
// Kernel 0: pack Wq/Wk/Wv/Wg/Wb into Wcat[528][128] (row c = output channel).
// ---------------------------------------------------------------------------
__global__ void __launch_bounds__(128)
repack_kernel(const float* __restrict__ Wq, const float* __restrict__ Wk,
              const float* __restrict__ Wv, const float* __restrict__ Wb,
              const float* __restrict__ Wg, float* __restrict__ Wcat) {
  const int c = blockIdx.x;     // 0..527
  const int d = threadIdx.x;    // 0..127
  float val = 0.0f;
  if (c < 128)      val = Wq[(size_t)c * 128 + d];          // (h*32+u)*128+d
  else if (c < 256) val = Wk[(size_t)(c - 128) * 128 + d];
  else if (c < 384) val = Wv[(size_t)(c - 256) * 128 + d];
  else if (c < 512) val = Wg[(size_t)(c - 384) * 128 + d];
  else if (c < 516) val = Wb[(size_t)(c - 512) * 128 + d];  // Wb (H,1,128)
  Wcat[(size_t)c * 128 + d] = val;
}

// ---------------------------------------------------------------------------
// Kernel 1: P[E][528] = Z[E][128] @ Wcat^T, sigmoid(+bg) on g channels.
// One WG (64 threads = 2 waves) per 16-edge M tile; waves split 33 N tiles.
// ---------------------------------------------------------------------------
__global__ void __launch_bounds__(64)
proj_kernel(const float* __restrict__ Z, const float* __restrict__ Wcat,
            const float* __restrict__ bg, float* __restrict__ P) {
  __shared__ float zt[TILE_M * IN_DIM];   // 8 KB
  const int m0 = blockIdx.x * TILE_M;
  const int t  = threadIdx.x;

  // prefetch next tile's Z rows while we work on this one
  if (m0 + TILE_M < E_TOTAL)
    __builtin_prefetch(Z + (size_t)(m0 + TILE_M) * IN_DIM, 0, 1);

  // cooperative Z tile load: 512 float4, 8 per thread
  {
    const float4* src = reinterpret_cast<const float4*>(Z + (size_t)m0 * IN_DIM);
    float4*       dst = reinterpret_cast<float4*>(zt);
#pragma unroll
    for (int i = 0; i < 8; ++i) dst[t + 64 * i] = src[t + 64 * i];
  }
  __syncthreads();

  const int lane = t & 31;
  const int wave = t >> 5;
  const int n    = lane & 15;    // A row (M) and B row (N) index
  const int half = lane >> 4;    // K sub-block select

  const float* arow = zt + (size_t)n * IN_DIM + 2 * half;

  for (int nt = wave; nt < NTILE_P; nt += 2) {
    const int c0 = nt * 16;
    const float* wrow = Wcat + (size_t)(c0 + n) * IN_DIM + 2 * half;

    v8f acc = {0.f, 0.f, 0.f, 0.f, 0.f, 0.f, 0.f, 0.f};
#pragma unroll 4
    for (int kk = 0; kk < IN_DIM / 4; ++kk) {
      v2f a = *reinterpret_cast<const v2f*>(arow + 4 * kk);
      v2f b = *reinterpret_cast<const v2f*>(wrow + 4 * kk);
      acc = __builtin_amdgcn_wmma_f32_16x16x4_f32(false, a, false, b,
                                                  (short)0, acc, false, false);
    }

    // g channels (384..511): sigmoid(x + bg[h*32+u]); bg flat index = c-384
    if (nt >= 24 && nt < 32) {
      const float bias = bg[c0 + n - 384];
#pragma unroll
      for (int r = 0; r < 8; ++r) {
        float x = acc[r] + bias;
        acc[r] = fast_rcp(1.0f + __expf(-x));   // v_rcp_f32, not IEEE divide
      }
    }

#pragma unroll
    for (int r = 0; r < 8; ++r) {
      const int row = m0 + r + 8 * half;
      P[(size_t)row * NCH + c0 + n] = acc[r];
    }
  }
}

// ---------------------------------------------------------------------------
// Kernel 2: per 16-edge tile: gather+softmax+att per (edge,head) thread,
// y = g*att staged in LDS, then y @ Wout^T + bout via WMMA into d_out.
// ---------------------------------------------------------------------------
__global__ void __launch_bounds__(64)
attn_kernel(const float* __restrict__ P, const long long* __restrict__ klist,
            const float* __restrict__ Wout, const float* __restrict__ bout,
            float* __restrict__ out) {
  __shared__ float ylds[TILE_M * OUT_DIM];   // 8 KB
  const int m0 = blockIdx.x * TILE_M;
  const int t  = threadIdx.x;
  const int el = t & 15;        // edge within tile
  const int h  = t >> 4;        // head
  const int e  = m0 + el;

  // ---- phase A: attention for (e, h) ----
  const float scale = 0.17677669529663687f;  // 1/sqrt(32)
  float4 q4[8], att4[8];
  {
    const float4* qp = reinterpret_cast<const float4*>(P + (size_t)e * NCH + h * HIDC);
#pragma unroll
    for (int i = 0; i < 8; ++i) {
      q4[i]   = qp[i];
      att4[i] = make_float4(0.f, 0.f, 0.f, 0.f);
    }
  }
  const long long* kio = klist + (size_t)e * (2 * MAXK);

  float sc[MAXK];
  int   nb[MAXK];
#pragma unroll
  for (int k = 0; k < MAXK; ++k) {
    const int iv = (int)kio[k];
    const int jv = (int)kio[MAXK + k];
    float s = 0.0f;                       // masked entries score exactly 0
    if (iv >= 0) {
      const float4* kp =
          reinterpret_cast<const float4*>(P + (size_t)iv * NCH + 128 + h * HIDC);
#pragma unroll
      for (int d = 0; d < 8; ++d) {
        const float4 kv = kp[d];
        s += q4[d].x * kv.x + q4[d].y * kv.y + q4[d].z * kv.z + q4[d].w * kv.w;
      }
      s = s * scale + P[(size_t)jv * NCH + 512 + h];   // + b[jj]
    }
    sc[k] = s;
    nb[k] = iv;
  }

  float mx = sc[0];
#pragma unroll
  for (int k = 1; k < MAXK; ++k) mx = fmaxf(mx, sc[k]);
  float sum = 0.0f;
#pragma unroll
  for (int k = 0; k < MAXK; ++k) { sc[k] = __expf(sc[k] - mx); sum += sc[k]; }
  const float inv = fast_rcp(sum);        // v_rcp_f32

#pragma unroll
  for (int k = 0; k < MAXK; ++k) {
    const int iv = nb[k];
    if (iv >= 0) {
      const float a = sc[k] * inv;
      const float4* vp =
          reinterpret_cast<const float4*>(P + (size_t)iv * NCH + 256 + h * HIDC);
#pragma unroll
      for (int d = 0; d < 8; ++d) {
        const float4 vv = vp[d];
        att4[d].x += a * vv.x; att4[d].y += a * vv.y;
        att4[d].z += a * vv.z; att4[d].w += a * vv.w;
      }
    }
  }

  // y = g * att -> LDS
  {
    const float4* gp =
        reinterpret_cast<const float4*>(P + (size_t)e * NCH + 384 + h * HIDC);
    float4* yrow = reinterpret_cast<float4*>(ylds + el * OUT_DIM + h * HIDC);
#pragma unroll
    for (int d = 0; d < 8; ++d) {
      const float4 g = gp[d];
      float4 y;
      y.x = g.x * att4[d].x; y.y = g.y * att4[d].y;
      y.z = g.z * att4[d].z; y.w = g.w * att4[d].w;
      yrow[d] = y;
    }
  }
  __syncthreads();

  // ---- phase B: out[m0..m0+15][:] = y @ Wout^T + bout (WMMA) ----
  const int lane = t & 31;
  const int wave = t >> 5;
  const int n    = lane & 15;
  const int half = lane >> 4;
  const float* arow = ylds + (size_t)n * OUT_DIM + 2 * half;

  for (int nt = wave; nt < OUT_DIM / 16; nt += 2) {
    const int c0 = nt * 16;
    const float* wrow = Wout + (size_t)(c0 + n) * OUT_DIM + 2 * half;

    v8f acc = {0.f, 0.f, 0.f, 0.f, 0.f, 0.f, 0.f, 0.f};
#pragma unroll 4
    for (int kk = 0; kk < OUT_DIM / 4; ++kk) {
      v2f a = *reinterpret_cast<const v2f*>(arow + 4 * kk);
      v2f b = *reinterpret_cast<const v2f*>(wrow + 4 * kk);
      acc = __builtin_amdgcn_wmma_f32_16x16x4_f32(false, a, false, b,
                                                  (short)0, acc, false, false);
    }

    const float bo = bout[c0 + n];
#pragma unroll
    for (int r = 0; r < 8; ++r) {
      const int row = m0 + r + 8 * half;
      out[(size_t)row * OUT_DIM + c0 + n] = acc[r] + bo;
    }
  }
}

// ---------------------------------------------------------------------------
extern "C" void kernel_launch(void* const* d_in, const int* in_sizes, int n_in,
                              void* d_out, int out_size, void* d_ws, size_t ws_size,
                              hipStream_t stream) {
  const float*     Z     = (const float*)d_in[0];
  const long long* klist = (const long long*)d_in[1];
  const float*     Wq    = (const float*)d_in[2];
  const float*     Wk    = (const float*)d_in[3];
  const float*     Wv    = (const float*)d_in[4];
  const float*     Wb    = (const float*)d_in[5];
  const float*     Wg    = (const float*)d_in[6];
  const float*     bg    = (const float*)d_in[7];
  const float*     Wout  = (const float*)d_in[8];
  const float*     bout  = (const float*)d_in[9];
  float*           outp  = (float*)d_out;

  float* wsf  = (float*)d_ws;
  float* Wcat = wsf;                        // 528*128 floats (270 KB)
  float* P    = wsf + (size_t)NCH * IN_DIM; // E*528 floats (~422 MB)

  repack_kernel<<<dim3(NCH), dim3(128), 0, stream>>>(Wq, Wk, Wv, Wb, Wg, Wcat);
  proj_kernel<<<dim3(NBLK), dim3(64), 0, stream>>>(Z, Wcat, bg, P);
  attn_kernel<<<dim3(NBLK), dim3(64), 0, stream>>>(P, klist, Wout, bout, outp);
}